// GIN_7730941133135
// MI455X (gfx1250) — compile-verified
//
#include <hip/hip_runtime.h>

#define NN 50000
#define NPAD 50048            // ceil(NN/128)*128 -> full waves everywhere, EXEC all-1s
#define NE 1600000
#define NFEAT 12
#define DIM 128
#define NCLASS 6

typedef float v2f __attribute__((ext_vector_type(2)));
typedef float v8f __attribute__((ext_vector_type(8)));

// ---------------- float4 copy (seeds agg = h, fusing the "+x" self term)
__global__ void copy_f4(const float* __restrict__ src, float* __restrict__ dst, int n4) {
    int i = blockIdx.x * blockDim.x + threadIdx.x;
    int stride = gridDim.x * blockDim.x;
    for (; i < n4; i += stride)
        ((float4*)dst)[i] = ((const float4*)src)[i];
}

// ---------------- edge scatter-add, 12 features (layer 1 input)
__global__ void agg_edges_12(const float* __restrict__ x, const int* __restrict__ ei,
                             float* __restrict__ agg) {
    long long i = (long long)blockIdx.x * blockDim.x + threadIdx.x;
    long long stride = (long long)gridDim.x * blockDim.x;
    const long long total = (long long)NE * NFEAT;
    for (; i < total; i += stride) {
        int e = (int)(i / NFEAT);
        int f = (int)(i % NFEAT);
        int s = ei[e];
        int d = ei[NE + e];
        atomicAdd(&agg[(long long)d * NFEAT + f], x[(long long)s * NFEAT + f]);
    }
}

// ---------------- edge scatter-add, 128 features: one wave per edge, float4 gather
__global__ void agg_edges_128(const float* __restrict__ h, const int* __restrict__ ei,
                              float* __restrict__ agg) {
    long long gid = (long long)blockIdx.x * blockDim.x + threadIdx.x;
    int lane = (int)(gid & 31);
    long long w  = gid >> 5;
    long long nw = ((long long)gridDim.x * blockDim.x) >> 5;
    for (long long e = w; e < NE; e += nw) {
        int s = ei[e];
        int d = ei[NE + e];
        float4 v = *(const float4*)(h + (long long)s * DIM + lane * 4);
        float* dp = agg + (long long)d * DIM + lane * 4;
        atomicAdd(dp + 0, v.x);
        atomicAdd(dp + 1, v.y);
        atomicAdd(dp + 2, v.z);
        atomicAdd(dp + 3, v.w);
    }
}

// ---------------- WMMA GEMM: out[NPAD x 128] = epi(A[NPAD x K] @ W[K x 128] + bias)
// epi = ReLU always, optional BatchNorm after ReLU.
// Block = 256 threads = 8 waves; each wave owns 16 rows x all 128 cols via
// 8 independent accumulators (breaks the WMMA D->C dependency chain and cuts
// A-fragment VMEM to 1/8 per WMMA). Weight panel (K x 128) staged in LDS.
__global__ __launch_bounds__(256)
void gemm_relu_bn(const float* __restrict__ A, const float* __restrict__ W,
                  const float* __restrict__ bias,
                  const float* __restrict__ gamma, const float* __restrict__ beta,
                  const float* __restrict__ mean, const float* __restrict__ var,
                  float* __restrict__ out, int K, int doBN) {
    extern __shared__ float wlds[];
    const int tid = threadIdx.x;
    for (int i = tid * 4; i < K * DIM; i += 256 * 4)
        *(float4*)(wlds + i) = *(const float4*)(W + i);
    __syncthreads();

    const int lane = tid & 31;
    const int wave = tid >> 5;                 // 0..7 -> 16-row strip within block
    const int m    = lane & 15;
    const int hf   = lane >> 4;                // 0/1: K-half within fragment
    const int r0   = blockIdx.x * 128 + wave * 16;

    const float* Arow = A + (long long)(r0 + m) * K;  // lane holds row r0+m

    v8f acc[8];
#pragma unroll
    for (int c = 0; c < 8; ++c) acc[c] = (v8f){0.f,0.f,0.f,0.f,0.f,0.f,0.f,0.f};

    for (int k0 = 0; k0 < K; k0 += 4) {
        // A 16x4 frag: VGPR0 = K{0|2}, VGPR1 = K{1|3} (lanes 0-15 | 16-31)
        float2 av = *(const float2*)(Arow + k0 + 2 * hf);
        v2f a; a[0] = av.x; a[1] = av.y;
        const float* Bk = wlds + (k0 + 2 * hf) * DIM + m;
#pragma unroll
        for (int c = 0; c < 8; ++c) {
            // B 4x16 frag for column tile c: row = K index, N = lane&15
            v2f b;
            b[0] = Bk[c * 16];
            b[1] = Bk[c * 16 + DIM];
            acc[c] = __builtin_amdgcn_wmma_f32_16x16x4_f32(
                         false, a, false, b, (short)0, acc[c], false, false);
        }
    }

#pragma unroll
    for (int c = 0; c < 8; ++c) {
        const int col = c * 16 + m;
        const float bv = bias[col];
        float sc = 1.f, mu = 0.f, bt = 0.f;
        if (doBN) {
            sc = gamma[col] * rsqrtf(var[col] + 1e-5f);
            mu = mean[col];
            bt = beta[col];
        }
#pragma unroll
        for (int r = 0; r < 8; ++r) {
            // C/D layout: VGPR r, lanes 0-15 -> row r; lanes 16-31 -> row r+8
            float v = acc[c][r] + bv;
            v = fmaxf(v, 0.f);
            if (doBN) v = (v - mu) * sc + bt;
            out[(long long)(r0 + r + 8 * hf) * DIM + col] = v;
        }
    }
}

// ---------------- final head: out[N x 6] = t @ fc2_w + fc2_b (tiny, VALU)
__global__ void head_fc2(const float* __restrict__ t, const float* __restrict__ w,
                         const float* __restrict__ b, float* __restrict__ out) {
    int i = blockIdx.x * blockDim.x + threadIdx.x;
    int stride = gridDim.x * blockDim.x;
    const int total = NN * NCLASS;
    for (; i < total; i += stride) {
        int n = i / NCLASS, c = i % NCLASS;
        const float* tr = t + (long long)n * DIM;
        float s = b[c];
#pragma unroll 4
        for (int k = 0; k < DIM; ++k) s += tr[k] * w[k * NCLASS + c];
        out[i] = s;
    }
}

extern "C" void kernel_launch(void* const* d_in, const int* in_sizes, int n_in,
                              void* d_out, int out_size, void* d_ws, size_t ws_size,
                              hipStream_t stream) {
    const float* x      = (const float*)d_in[0];
    const int*   ei     = (const int*)d_in[1];
    const float* w1_0   = (const float*)d_in[2];
    const float* b1_0   = (const float*)d_in[3];
    const float* w2_0   = (const float*)d_in[4];
    const float* b2_0   = (const float*)d_in[5];
    const float* ws1    = (const float*)d_in[6];
    const float* bs1    = (const float*)d_in[7];
    const float* ws2    = (const float*)d_in[8];
    const float* bs2    = (const float*)d_in[9];
    const float* gammas = (const float*)d_in[10];
    const float* betas  = (const float*)d_in[11];
    const float* means  = (const float*)d_in[12];
    const float* vars   = (const float*)d_in[13];
    const float* fc1_w  = (const float*)d_in[14];
    const float* fc1_b  = (const float*)d_in[15];
    const float* fc2_w  = (const float*)d_in[16];
    const float* fc2_b  = (const float*)d_in[17];
    float* out = (float*)d_out;

    char* ws = (char*)d_ws;
    float* t     = (float*)(ws);                                    // NPAD*128
    float* h     = (float*)(ws + (size_t)NPAD * DIM * 4);           // NPAD*128
    float* agg   = (float*)(ws + (size_t)NPAD * DIM * 8);           // NPAD*128
    float* agg12 = (float*)(ws + (size_t)NPAD * DIM * 12);          // NPAD*12

    const int nBlk = NPAD / 128;                                    // 391 full blocks
    const size_t lds128 = (size_t)DIM * DIM * sizeof(float);        // 64 KB
    const size_t lds12  = (size_t)NFEAT * DIM * sizeof(float);      // 6 KB

    // ---- layer 1: agg12 = x + scatter_sum(x[src] -> dst)
    copy_f4<<<1024, 256, 0, stream>>>(x, agg12, NN * NFEAT / 4);
    agg_edges_12<<<6144, 256, 0, stream>>>(x, ei, agg12);
    gemm_relu_bn<<<nBlk, 256, lds12, stream>>>(
        agg12, w1_0, b1_0, nullptr, nullptr, nullptr, nullptr, t, NFEAT, 0);
    gemm_relu_bn<<<nBlk, 256, lds128, stream>>>(
        t, w2_0, b2_0, gammas, betas, means, vars, h, DIM, 1);

    // ---- layers 2-5
    for (int i = 0; i < 4; ++i) {
        copy_f4<<<2048, 256, 0, stream>>>(h, agg, NN * DIM / 4);
        agg_edges_128<<<6144, 256, 0, stream>>>(h, ei, agg);
        gemm_relu_bn<<<nBlk, 256, lds128, stream>>>(
            agg, ws1 + (size_t)i * DIM * DIM, bs1 + i * DIM,
            nullptr, nullptr, nullptr, nullptr, t, DIM, 0);
        gemm_relu_bn<<<nBlk, 256, lds128, stream>>>(
            t, ws2 + (size_t)i * DIM * DIM, bs2 + i * DIM,
            gammas + (i + 1) * DIM, betas + (i + 1) * DIM,
            means + (i + 1) * DIM, vars + (i + 1) * DIM, h, DIM, 1);
    }

    // ---- heads
    gemm_relu_bn<<<nBlk, 256, lds128, stream>>>(
        h, fc1_w, fc1_b, nullptr, nullptr, nullptr, nullptr, t, DIM, 0);
    head_fc2<<<2048, 256, 0, stream>>>(t, fc2_w, fc2_b, out);
}